// StackAugmentedRNN_28432683500169
// MI455X (gfx1250) — compile-verified
//
#include <hip/hip_runtime.h>
#include <hip/hip_bf16.h>
#include <math.h>

// Sizes from the reference
#define V_SZ 45
#define H_SZ 2048
#define W_SZ 128
#define D_SZ 200
#define KIH  (H_SZ + W_SZ)   // 2176
#define N3H  (3 * H_SZ)      // 6144

typedef float v2f __attribute__((ext_vector_type(2)));
typedef float v8f __attribute__((ext_vector_type(8)));

// ---------------------------------------------------------------------------
// Kernel 1: ctrl logits (rows 0..2, no activation) and stack_input rows
// (rows 3..130, tanh). One wave (32 lanes) per output row, K = 2048.
// ---------------------------------------------------------------------------
__global__ __launch_bounds__(32) void k_ctrl_sin(
    const float* __restrict__ h,
    const float* __restrict__ ctrl_W, const float* __restrict__ ctrl_b,
    const float* __restrict__ sin_W,  const float* __restrict__ sin_b,
    float* __restrict__ logits, float* __restrict__ stack_in)
{
    const int row = blockIdx.x;            // 0..130
    const float* Wr;
    const float* bp;
    float* outp;
    int orow;
    bool doTanh;
    if (row < 3) { Wr = ctrl_W; bp = ctrl_b; outp = logits;   orow = row;     doTanh = false; }
    else         { Wr = sin_W;  bp = sin_b;  outp = stack_in; orow = row - 3; doTanh = true;  }
    const float* wr = Wr + (size_t)orow * H_SZ;

    const int lane = threadIdx.x;
    float s = 0.f;
#pragma unroll 4
    for (int k = lane * 4; k < H_SZ; k += 32 * 4) {
        float4 wv = *(const float4*)(wr + k);
        float4 hv = *(const float4*)(h  + k);
        s += wv.x * hv.x + wv.y * hv.y + wv.z * hv.z + wv.w * hv.w;
    }
    for (int off = 16; off > 0; off >>= 1) s += __shfl_down(s, off, 32);
    if (lane == 0) {
        s += bp[orow];
        outp[orow] = doTanh ? tanhf(s) : s;
    }
}

// ---------------------------------------------------------------------------
// Kernel 2: softmax(3) inline, stack blend (25600 elems) -> out_stack,
// and assemble x = [emb[inp] (2048) | new_stack[0] (128)] into ws.
// grid 108 x 256 = 27648 threads = 25600 + 2048 exactly.
// ---------------------------------------------------------------------------
__global__ __launch_bounds__(256) void k_stack(
    const float* __restrict__ stack, const float* __restrict__ logits,
    const float* __restrict__ stack_in, const float* __restrict__ emb,
    const int* __restrict__ inp,
    float* __restrict__ out_stack, float* __restrict__ xbuf)
{
    const float l0 = logits[0], l1 = logits[1], l2 = logits[2];
    const float m  = fmaxf(l0, fmaxf(l1, l2));
    const float e0 = expf(l0 - m), e1 = expf(l1 - m), e2 = expf(l2 - m);
    const float inv = 1.f / (e0 + e1 + e2);
    const float c0 = e0 * inv, c1 = e1 * inv, c2 = e2 * inv;

    const int idx = blockIdx.x * blockDim.x + threadIdx.x;
    if (idx < D_SZ * W_SZ) {
        const int d = idx >> 7;          // /128
        const int w = idx & 127;
        const float cur  = stack[idx];
        const float up   = (d == 0)        ? stack_in[w]     : stack[idx - W_SZ];
        const float down = (d == D_SZ - 1) ? 0.f             : stack[idx + W_SZ];
        const float v = c2 * cur + c0 * up + c1 * down;
        out_stack[idx] = v;
        if (d == 0) xbuf[H_SZ + w] = v;   // stack_top
    } else {
        const int j = idx - D_SZ * W_SZ;  // 0..2047
        xbuf[j] = emb[(size_t)inp[0] * H_SZ + j];
    }
}

// ---------------------------------------------------------------------------
// Kernel 3: big memory-bound GEMV via V_WMMA_F32_16X16X4_F32, templated on
// the compile-time reduction length K so the k-loop trip count is constant
// and unrolls by 4 (more loads in flight per wave, less branch overhead).
//
// Block = 128 threads = 4 waves. Each block owns a 16-row tile of W; waves
// split K 4-ways. B operand = W tile (lane n loads b64 at row n, k + hi*2,
// matching the ISA 32-bit B layout). A operand = x in row M=0 only (lanes 0
// and 16 carry data, other A rows zeroed). y[n] lands in c[0] of lanes 0..15.
// Dual accumulators break the WMMA->WMMA RAW chain; LDS reduces 4 partials.
// ---------------------------------------------------------------------------
template <int K>
__global__ __launch_bounds__(128) void k_gemv_wmma(
    const float* __restrict__ W, const float* __restrict__ xv,
    const float* __restrict__ bias, float* __restrict__ out)
{
    const int n_base = blockIdx.x * 16;
    const int lane   = threadIdx.x & 31;
    const int wave   = threadIdx.x >> 5;     // 0..3
    const int laneN  = lane & 15;
    const int hi     = lane >> 4;            // 0 or 1
    const bool isRow0 = (laneN == 0);        // lanes 0 and 16 carry A row M=0

    constexpr int kPerWave = K >> 2;         // 544 or 512 (both % 8 == 0)
    const int kb = wave * kPerWave;

    const float* __restrict__ wrow = W + (size_t)(n_base + laneN) * K + hi * 2;

    v8f c0 = {0.f, 0.f, 0.f, 0.f, 0.f, 0.f, 0.f, 0.f};
    v8f c1 = {0.f, 0.f, 0.f, 0.f, 0.f, 0.f, 0.f, 0.f};

#pragma unroll 4
    for (int kk = 0; kk < kPerWave; kk += 8) {
        const int k = kb + kk;
        v2f b0 = *(const v2f*)(wrow + k);
        v2f b1 = *(const v2f*)(wrow + k + 4);
        v2f xa = *(const v2f*)(xv + k + hi * 2);
        v2f xb = *(const v2f*)(xv + k + 4 + hi * 2);
        v2f a0, a1;
        a0.x = isRow0 ? xa.x : 0.f;  a0.y = isRow0 ? xa.y : 0.f;
        a1.x = isRow0 ? xb.x : 0.f;  a1.y = isRow0 ? xb.y : 0.f;
        c0 = __builtin_amdgcn_wmma_f32_16x16x4_f32(false, a0, false, b0,
                                                   (short)0, c0, false, false);
        c1 = __builtin_amdgcn_wmma_f32_16x16x4_f32(false, a1, false, b1,
                                                   (short)0, c1, false, false);
    }

    __shared__ float red[4][16];
    if (lane < 16) red[wave][lane] = c0[0] + c1[0];   // M=0, N=lane
    __syncthreads();
    if (threadIdx.x < 16) {
        const int n = n_base + threadIdx.x;
        const float s = red[0][threadIdx.x] + red[1][threadIdx.x] +
                        red[2][threadIdx.x] + red[3][threadIdx.x];
        out[n] = s + bias[n];
    }
}

// ---------------------------------------------------------------------------
// Kernel 4: GRU gate elementwise. 2048 outputs.
// ---------------------------------------------------------------------------
__global__ __launch_bounds__(256) void k_gru(
    const float* __restrict__ gi, const float* __restrict__ gh,
    const float* __restrict__ h,
    float* __restrict__ out_h, float* __restrict__ hnew_ws)
{
    const int i = blockIdx.x * blockDim.x + threadIdx.x;
    if (i >= H_SZ) return;
    const float ir  = gi[i],           hr = gh[i];
    const float iz  = gi[H_SZ + i],    hz = gh[H_SZ + i];
    const float in_ = gi[2*H_SZ + i],  hn = gh[2*H_SZ + i];
    const float r = 1.f / (1.f + expf(-(ir + hr)));
    const float z = 1.f / (1.f + expf(-(iz + hz)));
    const float n = tanhf(in_ + r * hn);
    const float hv = h[i];
    const float o = (1.f - z) * n + z * hv;
    out_h[i]   = o;
    hnew_ws[i] = o;
}

// ---------------------------------------------------------------------------
// Kernel 5: output = dec_W (45 x 2048) * h_new + dec_b. One wave per row.
// ---------------------------------------------------------------------------
__global__ __launch_bounds__(32) void k_dec(
    const float* __restrict__ hnew, const float* __restrict__ dec_W,
    const float* __restrict__ dec_b, float* __restrict__ out_y)
{
    const int row = blockIdx.x;            // 0..44
    const float* wr = dec_W + (size_t)row * H_SZ;
    const int lane = threadIdx.x;
    float s = 0.f;
#pragma unroll 4
    for (int k = lane * 4; k < H_SZ; k += 32 * 4) {
        float4 wv = *(const float4*)(wr + k);
        float4 hv = *(const float4*)(hnew + k);
        s += wv.x * hv.x + wv.y * hv.y + wv.z * hv.z + wv.w * hv.w;
    }
    for (int off = 16; off > 0; off >>= 1) s += __shfl_down(s, off, 32);
    if (lane == 0) out_y[row] = s + dec_b[row];
}

// ---------------------------------------------------------------------------
extern "C" void kernel_launch(void* const* d_in, const int* in_sizes, int n_in,
                              void* d_out, int out_size, void* d_ws, size_t ws_size,
                              hipStream_t stream)
{
    (void)in_sizes; (void)n_in; (void)out_size; (void)ws_size;

    const int*   inp    = (const int*)  d_in[0];
    const float* hidden = (const float*)d_in[1];   // (1,1,2048)
    const float* stack  = (const float*)d_in[2];   // (1,200,128)
    const float* emb    = (const float*)d_in[3];   // (45,2048)
    const float* ctrl_W = (const float*)d_in[4];   // (3,2048)
    const float* ctrl_b = (const float*)d_in[5];
    const float* sin_W  = (const float*)d_in[6];   // (128,2048)
    const float* sin_b  = (const float*)d_in[7];
    const float* w_ih   = (const float*)d_in[8];   // (6144,2176)
    const float* w_hh   = (const float*)d_in[9];   // (6144,2048)
    const float* b_ih   = (const float*)d_in[10];
    const float* b_hh   = (const float*)d_in[11];
    const float* dec_W  = (const float*)d_in[12];  // (45,2048)
    const float* dec_b  = (const float*)d_in[13];

    float* out       = (float*)d_out;
    float* out_y     = out;                    // 45
    float* out_h     = out + V_SZ;             // 2048
    float* out_stack = out + V_SZ + H_SZ;      // 25600

    float* ws       = (float*)d_ws;
    float* logits   = ws;            // 3
    float* stack_in = ws + 16;       // 128
    float* xbuf     = ws + 160;      // 2176
    float* gi       = ws + 2336;     // 6144
    float* gh       = ws + 8480;     // 6144
    float* hnew     = ws + 14624;    // 2048   (total ~65 KB)

    // 1) controls logits + tanh(stack input row)
    k_ctrl_sin<<<131, 32, 0, stream>>>(hidden, ctrl_W, ctrl_b, sin_W, sin_b,
                                       logits, stack_in);
    // 2) softmax + stack blend + x assembly
    k_stack<<<108, 256, 0, stream>>>(stack, logits, stack_in, emb, inp,
                                     out_stack, xbuf);
    // 3) big HBM-bound GEMVs via f32 WMMA (gi then gh; independent blocks)
    k_gemv_wmma<KIH><<<N3H / 16, 128, 0, stream>>>(w_ih, xbuf,  b_ih, gi);
    k_gemv_wmma<H_SZ><<<N3H / 16, 128, 0, stream>>>(w_hh, hidden, b_hh, gh);
    // 4) GRU gates
    k_gru<<<(H_SZ + 255) / 256, 256, 0, stream>>>(gi, gh, hidden, out_h, hnew);
    // 5) decode
    k_dec<<<V_SZ, 32, 0, stream>>>(hnew, dec_W, dec_b, out_y);
}